// ReftModule_23012434772608
// MI455X (gfx1250) — compile-verified
//
#include <hip/hip_runtime.h>

typedef float v2f __attribute__((ext_vector_type(2)));
typedef float v4f __attribute__((ext_vector_type(4)));
typedef float v8f __attribute__((ext_vector_type(8)));

#define B_  4
#define L_  4096
#define D_  4096
#define R_  32

// ---------------------------------------------------------------------------
// Edited-row addressing: unit 0 = prefix (offsets[b]+j), unit 1 = suffix
// (offsets[b]+seqlens[b]-4+j).  i in [0,16): b = i>>2, j = i&3.
// ---------------------------------------------------------------------------
__device__ __forceinline__ size_t edited_row_base(int unit, int i,
                                                  const int* __restrict__ offsets,
                                                  const int* __restrict__ seqlens) {
    int b = i >> 2;
    int j = i & 3;
    int pos = offsets[b] + j + (unit ? (seqlens[b] - 4) : 0);
    return ((size_t)b * L_ + (size_t)pos) * (size_t)D_;
}

// ---------------------------------------------------------------------------
// Kernel 1: bulk streaming copy, 128-bit nontemporal loads/stores.
// 256 MB in + 256 MB out -> HBM-bound (~22 us at 23.3 TB/s).
// ---------------------------------------------------------------------------
__global__ void bulk_copy_b128(const v4f* __restrict__ in, v4f* __restrict__ out,
                               size_t n4) {
    size_t i      = (size_t)blockIdx.x * blockDim.x + threadIdx.x;
    size_t stride = (size_t)gridDim.x * blockDim.x;
    for (; i < n4; i += stride) {
        v4f v = __builtin_nontemporal_load(&in[i]);
        __builtin_nontemporal_store(v, &out[i]);
    }
}

// ---------------------------------------------------------------------------
// Kernel 2 (phase 1): S[unit][16][32] = relu(X Wsrc^T + b) - X Wproj^T.
// Grid: (rtile 0..1, unit 0..1). 8 waves split K=4096; per-wave WMMA chains,
// cross-wave reduction via LDS ds_add_f32 atomics.
// ---------------------------------------------------------------------------
__global__ void reft_phase1(const float* __restrict__ X,
                            const float* __restrict__ Wsrc_p, const float* __restrict__ bsrc_p,
                            const float* __restrict__ Wproj_p,
                            const float* __restrict__ Wsrc_s, const float* __restrict__ bsrc_s,
                            const float* __restrict__ Wproj_s,
                            const int* __restrict__ offsets, const int* __restrict__ seqlens,
                            float* __restrict__ Sbuf /* [2][16][32] */) {
    const int unit  = blockIdx.y;
    const int rtile = blockIdx.x;

    const float* __restrict__ Wsrc  = unit ? Wsrc_s  : Wsrc_p;
    const float* __restrict__ Wproj = unit ? Wproj_s : Wproj_p;
    const float* __restrict__ bsrc  = unit ? bsrc_s  : bsrc_p;

    const int tid  = threadIdx.x;
    const int lane = tid & 31;
    const int wave = tid >> 5;          // 0..7

    const int m  = lane & 15;           // A-matrix row for this lane
    const int kh = (lane >> 4) << 1;    // K sub-offset (0 or 2)
    const int r  = rtile * 16 + (lane & 15);  // B-matrix column (rank index)

    const float* __restrict__ xrow = X + edited_row_base(unit, m, offsets, seqlens);
    const float* __restrict__ wsr  = Wsrc  + (size_t)r * D_;
    const float* __restrict__ wpr  = Wproj + (size_t)r * D_;

    v8f acc_s = {};
    v8f acc_p = {};

    const int k_begin = wave * (D_ / 8);
    const int k_end   = k_begin + (D_ / 8);
    for (int k = k_begin; k < k_end; k += 4) {
        v2f a, bs, bp;
        a.x  = xrow[k + kh];     a.y  = xrow[k + kh + 1];
        bs.x = wsr[k + kh];      bs.y = wsr[k + kh + 1];   // B[k][n] = Wsrc[r][k]
        bp.x = wpr[k + kh];      bp.y = wpr[k + kh + 1];
        acc_s = __builtin_amdgcn_wmma_f32_16x16x4_f32(false, a, false, bs,
                                                      (short)0, acc_s, false, false);
        acc_p = __builtin_amdgcn_wmma_f32_16x16x4_f32(false, a, false, bp,
                                                      (short)0, acc_p, false, false);
    }

    // Cross-wave reduction of the 16x16 partial tiles (per-lane 8 accumulators).
    __shared__ float red[2][32][8];
    for (int t = tid; t < 2 * 32 * 8; t += blockDim.x) ((float*)red)[t] = 0.0f;
    __syncthreads();
#pragma unroll
    for (int v = 0; v < 8; ++v) {
        atomicAdd(&red[0][lane][v], acc_s[v]);
        atomicAdd(&red[1][lane][v], acc_p[v]);
    }
    __syncthreads();

    if (tid < 32) {
        const float bias = bsrc[r];
#pragma unroll
        for (int v = 0; v < 8; ++v) {
            int mm = v + ((lane >> 4) << 3);   // C/D layout: m = vgpr + 8*(lane>=16)
            float s = red[0][lane][v] + bias;
            s = s > 0.0f ? s : 0.0f;
            s -= red[1][lane][v];
            Sbuf[((size_t)unit * 16 + mm) * 32 + r] = s;
        }
    }
}

// ---------------------------------------------------------------------------
// Kernel 3 (phase 2): out[rows] = X + S * Wproj.  Each wave handles one
// 16-column N-tile: C seeded with gathered X tile, 8 chained WMMAs over K=32.
// ---------------------------------------------------------------------------
__global__ void reft_phase2(const float* __restrict__ X,
                            const float* __restrict__ Wproj_p,
                            const float* __restrict__ Wproj_s,
                            const int* __restrict__ offsets, const int* __restrict__ seqlens,
                            const float* __restrict__ Sbuf,
                            float* __restrict__ out) {
    const int unit  = blockIdx.y;
    const int tid   = threadIdx.x;
    const int lane  = tid & 31;
    const int wave  = tid >> 5;
    const int ntile = blockIdx.x * 8 + wave;   // 0..255
    const int n0    = ntile * 16;

    const float* __restrict__ Wproj = unit ? Wproj_s : Wproj_p;
    const float* __restrict__ S     = Sbuf + (size_t)unit * 16 * 32;

    const int m  = lane & 15;
    const int kh = (lane >> 4) << 1;
    const int nc = lane & 15;
    const int hi = lane >> 4;

    // A-matrix frags: S row m, K = 0..31 in steps of 4.
    const float* __restrict__ srow = S + m * 32;
    v2f a[8];
#pragma unroll
    for (int kk = 0; kk < 8; ++kk) {
        a[kk].x = srow[kk * 4 + kh];
        a[kk].y = srow[kk * 4 + kh + 1];
    }

    // C init = gathered X tile (rows = edited rows, cols n0..n0+15).
    v8f c;
#pragma unroll
    for (int v = 0; v < 8; ++v) {
        int mm = v + hi * 8;
        const float* xr = X + edited_row_base(unit, mm, offsets, seqlens);
        c[v] = xr[n0 + nc];
    }

#pragma unroll
    for (int kk = 0; kk < 8; ++kk) {
        v2f b;
        b.x = Wproj[(size_t)(kk * 4 + kh)     * D_ + n0 + nc];  // B[k][n] = Wproj[k][col]
        b.y = Wproj[(size_t)(kk * 4 + kh + 1) * D_ + n0 + nc];
        c = __builtin_amdgcn_wmma_f32_16x16x4_f32(false, a[kk], false, b,
                                                  (short)0, c, false, false);
    }

    // Scatter-store edited rows into out (overwrites bulk copy; stream-ordered).
#pragma unroll
    for (int v = 0; v < 8; ++v) {
        int mm = v + hi * 8;
        size_t base = edited_row_base(unit, mm, offsets, seqlens);
        out[base + n0 + nc] = c[v];
    }
}

// ---------------------------------------------------------------------------
extern "C" void kernel_launch(void* const* d_in, const int* in_sizes, int n_in,
                              void* d_out, int out_size, void* d_ws, size_t ws_size,
                              hipStream_t stream) {
    const float* hs  = (const float*)d_in[0];
    const float* Wsp = (const float*)d_in[1];
    const float* bsp = (const float*)d_in[2];
    const float* Wpp = (const float*)d_in[3];
    const float* Wss = (const float*)d_in[4];
    const float* bss = (const float*)d_in[5];
    const float* Wps = (const float*)d_in[6];
    const int*   off = (const int*)d_in[7];
    const int*   sl  = (const int*)d_in[8];
    float*       out = (float*)d_out;
    float*      Sbuf = (float*)d_ws;   // needs 2*16*32 floats = 4 KB

    const size_t n4 = (size_t)B_ * L_ * D_ / 4;   // 16,777,216 float4s

    bulk_copy_b128<<<4096, 256, 0, stream>>>((const v4f*)hs, (v4f*)out, n4);
    reft_phase1<<<dim3(2, 2), 256, 0, stream>>>(hs, Wsp, bsp, Wpp, Wss, bss, Wps,
                                                off, sl, Sbuf);
    reft_phase2<<<dim3(32, 2), 256, 0, stream>>>(hs, Wpp, Wps, off, sl, Sbuf, out);
}